// comb_69329362092380
// MI455X (gfx1250) — compile-verified
//
#include <hip/hip_runtime.h>
#include <cstring>
#include <cstdint>

// ======================================================================
// GNN forward for MI455X (gfx1250, wave32).
// Dense linears + per-edge message MLPs run on V_WMMA_F32_16X16X4_F32
// (exact fp32 math through the matrix pipe). Graph aggregation uses
// global float atomics; segment-max uses an order-preserving u32 encode
// so native atomicMax(u32) works for floats.
//
// All WMMA operand loads are branchless: addresses are clamped to valid
// ranges and results zeroed with cndmask, so the inner loop is pure
// {b64 load, cndmask, wmma} with EXEC all-1 throughout (WMMA requirement).
// Requires even inF (holds for all call sites: 10/16/26, +8 edge feats).
//
// Input order assumption: 8 top-level arrays in setup_inputs() insertion
// order, then params flattened as jax pytree leaves (dict keys sorted):
// enc, gat2, gat3, gat5, gc1, l1, l2, l3, lin, linxz, sage4, tc1, tc2,
// with each sub-dict's keys in ASCII-sorted order.
// ======================================================================

typedef __attribute__((ext_vector_type(2))) float v2f;
typedef __attribute__((ext_vector_type(8))) float v8f;

#define DEVF __device__ __forceinline__

// ---- order-preserving float<->u32 encoding (for atomicMax on floats) ----
DEVF unsigned encOrd(float f) {
  unsigned u = __float_as_uint(f);
  return (u & 0x80000000u) ? ~u : (u | 0x80000000u);
}
DEVF float decOrd(unsigned u) {
  u = (u & 0x80000000u) ? (u & 0x7fffffffu) : ~u;
  return __uint_as_float(u);
}
DEVF void atomicMaxEnc(float* a, float v) {
  atomicMax((unsigned int*)a, encOrd(v));
}
// encOrd(-inf) bit pattern (buffer init value for encoded max buffers)
#define ENC_NEG_INF_U 0x007FFFFFu

// ======================================================================
// Kernels
// ======================================================================

__global__ void k_fill(float* p, float v, int n) {
  int i = blockIdx.x * blockDim.x + threadIdx.x;
  if (i < n) p[i] = v;
}

// Y[n,outF] = act( X[n,inF] @ W[inF,outF] + bias + residual )
// One wave per 16-row tile; K tiled by 4 via V_WMMA_F32_16X16X4_F32.
// inF must be even (pair loads); outF <= 16.
__global__ void k_linear_wmma(const float* __restrict__ X, const float* __restrict__ W,
                              const float* __restrict__ B, const float* __restrict__ R,
                              float* __restrict__ Y, int n, int inF, int outF, int doRelu) {
  int gtid = blockIdx.x * blockDim.x + threadIdx.x;
  int wave = gtid >> 5;
  int lane = gtid & 31;
  int rowBase = wave * 16;
  if (rowBase >= n) return;                 // uniform per wave -> EXEC stays all-1 at WMMA
  int half = lane >> 4;                     // 0: lanes 0-15, 1: lanes 16-31
  int l = lane & 15;
  int row = rowBase + l;                    // A-matrix row for this lane
  bool rowOK = row < n;
  int rowC = rowOK ? row : 0;               // clamped, always valid
  int lc = (l < outF) ? l : 0;              // clamped B column
  v8f c = {0.f, 0.f, 0.f, 0.f, 0.f, 0.f, 0.f, 0.f};
  for (int kk = 0; kk < inF; kk += 4) {
    int k0 = kk + half * 2;                 // even; pair (k0,k0+1) never straddles a row
    bool kOK = k0 < inF;
    int k0c = kOK ? k0 : 0;
    // A: one unconditional 8B load from clamped address, then mask
    v2f av = *(const v2f*)(X + (size_t)rowC * inF + k0c);
    bool mA = rowOK & kOK;
    v2f a, bb;
    a.x = mA ? av.x : 0.f;
    a.y = mA ? av.y : 0.f;
    // B: two unconditional scalar loads from clamped indices, then mask
    float w0 = W[(size_t)k0c * outF + lc];
    float w1 = W[(size_t)(k0c + 1) * outF + lc];
    bool mB = kOK & (l < outF);
    bb.x = mB ? w0 : 0.f;
    bb.y = mB ? w1 : 0.f;
    c = __builtin_amdgcn_wmma_f32_16x16x4_f32(false, a, false, bb, (short)0, c, false, false);
  }
  if (l < outF) {
    float bias = B ? B[l] : 0.f;
#pragma unroll
    for (int j = 0; j < 8; ++j) {           // C: VGPR j = row j (lo half) / row j+8 (hi half)
      int r = rowBase + half * 8 + j;
      if (r < n) {
        float v = c[j] + bias;
        if (R) v += R[(size_t)r * outF + l];
        if (doRelu) v = fmaxf(v, 0.f);
        Y[(size_t)r * outF + l] = v;
      }
    }
  }
}

// per-edge message MLP: m = relu([x_src || e_attr] @ Wm + bm); agg[tgt] += m
// One wave per 16 edges; A rows gathered through src[] into WMMA layout.
// inF even -> the (k0,k0+1) pair never straddles the X/EA boundary.
__global__ void k_own_edge_wmma(const float* __restrict__ X, const float* __restrict__ EA,
                                const int* __restrict__ src, const int* __restrict__ tgt,
                                const float* __restrict__ Wm, const float* __restrict__ bm,
                                float* __restrict__ agg, int E, int inF) {
  int gtid = blockIdx.x * blockDim.x + threadIdx.x;
  int wave = gtid >> 5;
  int lane = gtid & 31;
  int eBase = wave * 16;
  if (eBase >= E) return;
  int half = lane >> 4;
  int l = lane & 15;
  int e = eBase + l;
  bool eOK = e < E;
  int eC = eOK ? e : (E - 1);               // clamped edge index (always valid)
  int s = src[eC];                          // unconditional, always valid
  int K = inF + 8;                          // even
  int lc = (l < 10) ? l : 0;
  v8f c = {0.f, 0.f, 0.f, 0.f, 0.f, 0.f, 0.f, 0.f};
  for (int kk = 0; kk < K; kk += 4) {
    int k0 = kk + half * 2;
    bool kOK = k0 < K;
    int k0c = kOK ? k0 : 0;
    // branchless source select: x-features vs edge-attr (pointer cndmask)
    const float* p = (k0c < inF) ? (X + (size_t)s * inF + k0c)
                                 : (EA + (size_t)eC * 8 + (k0c - inF));
    v2f av = *(const v2f*)p;                // one unconditional 8B load
    bool mA = eOK & kOK;
    v2f a, bb;
    a.x = mA ? av.x : 0.f;
    a.y = mA ? av.y : 0.f;
    float w0 = Wm[(size_t)k0c * 10 + lc];
    float w1 = Wm[(size_t)(k0c + 1) * 10 + lc];
    bool mB = kOK & (l < 10);
    bb.x = mB ? w0 : 0.f;
    bb.y = mB ? w1 : 0.f;
    c = __builtin_amdgcn_wmma_f32_16x16x4_f32(false, a, false, bb, (short)0, c, false, false);
  }
  if (l < 10) {
    float bias = bm[l];
#pragma unroll
    for (int j = 0; j < 8; ++j) {
      int ee = eBase + half * 8 + j;
      if (ee < E) {
        float v = fmaxf(c[j] + bias, 0.f);
        atomicAdd(&agg[(size_t)tgt[ee] * 10 + l], v);
      }
    }
  }
}

__global__ void k_deg(const int* __restrict__ tgt, float* cnt, int E) {
  int i = blockIdx.x * blockDim.x + threadIdx.x;
  if (i < E) atomicAdd(&cnt[tgt[i]], 1.f);
}

// TransformerConv edge score: e = <q[tgt], k[src]> / sqrt(10); seg-max
__global__ void k_tc_edge(const float* __restrict__ q, const float* __restrict__ k,
                          const int* __restrict__ src, const int* __restrict__ tgt,
                          float* ew, float* segM, int E) {
  int i = blockIdx.x * blockDim.x + threadIdx.x;
  if (i >= E) return;
  int a = src[i], b = tgt[i];
  float sacc = 0.f;
#pragma unroll
  for (int d = 0; d < 10; ++d) sacc += q[(size_t)b * 10 + d] * k[(size_t)a * 10 + d];
  sacc *= 0.31622776601683794f;  // 1/sqrt(10)
  ew[i] = sacc;
  atomicMaxEnc(&segM[b], sacc);
}

// GATv2 edge score: e = <att, leaky_relu(xl[src]+xr[tgt], 0.2)>; seg-max
__global__ void k_gat_edge(const float* __restrict__ xl, const float* __restrict__ xr,
                           const float* __restrict__ att,
                           const int* __restrict__ src, const int* __restrict__ tgt,
                           float* ew, float* segM, int E) {
  int i = blockIdx.x * blockDim.x + threadIdx.x;
  if (i >= E) return;
  int a = src[i], b = tgt[i];
  float sacc = 0.f;
#pragma unroll
  for (int d = 0; d < 10; ++d) {
    float t = xl[(size_t)a * 10 + d] + xr[(size_t)b * 10 + d];
    t = (t > 0.f) ? t : 0.2f * t;
    sacc += att[d] * t;
  }
  ew[i] = sacc;
  atomicMaxEnc(&segM[b], sacc);
}

// w = exp(e - m[tgt]); s[tgt] += w   (segM is encoded)
__global__ void k_exp_sum(const float* __restrict__ segM, float* ew, float* segS,
                          const int* __restrict__ tgt, int E) {
  int i = blockIdx.x * blockDim.x + threadIdx.x;
  if (i >= E) return;
  int b = tgt[i];
  float m = decOrd(__float_as_uint(segM[b]));
  float w = __expf(ew[i] - m);
  ew[i] = w;
  atomicAdd(&segS[b], w);
}

// a = w/(s+1e-16); agg[tgt] += a * V[src]
__global__ void k_soft_scatter(const float* __restrict__ ew, const float* __restrict__ segS,
                               const float* __restrict__ V,
                               const int* __restrict__ src, const int* __restrict__ tgt,
                               float* agg, int E) {
  int i = blockIdx.x * blockDim.x + threadIdx.x;
  if (i >= E) return;
  int s = src[i], t = tgt[i];
  float a = ew[i] / (segS[t] + 1e-16f);
#pragma unroll
  for (int d = 0; d < 10; ++d)
    atomicAdd(&agg[(size_t)t * 10 + d], a * V[(size_t)s * 10 + d]);
}

__global__ void k_sum_scatter(const float* __restrict__ M, const int* __restrict__ src,
                              const int* __restrict__ tgt, float* agg, int E) {
  int i = blockIdx.x * blockDim.x + threadIdx.x;
  if (i >= E) return;
  int s = src[i], t = tgt[i];
#pragma unroll
  for (int d = 0; d < 10; ++d)
    atomicAdd(&agg[(size_t)t * 10 + d], M[(size_t)s * 10 + d]);
}

__global__ void k_max_scatter(const float* __restrict__ X, const int* __restrict__ src,
                              const int* __restrict__ tgt, float* aggE, int E) {
  int i = blockIdx.x * blockDim.x + threadIdx.x;
  if (i >= E) return;
  int s = src[i], t = tgt[i];
#pragma unroll
  for (int d = 0; d < 10; ++d)
    atomicMaxEnc(&aggE[(size_t)t * 10 + d], X[(size_t)s * 10 + d]);
}

// decode encoded max buffer; empty segments (-inf) -> 0
__global__ void k_dec_sanitize(float* p, int n) {
  int i = blockIdx.x * blockDim.x + threadIdx.x;
  if (i >= n) return;
  float f = decOrd(__float_as_uint(p[i]));
  p[i] = (f > -1e37f) ? f : 0.f;
}

__global__ void k_add_relu(const float* A, const float* Bv, float* Y, int n, int doRelu) {
  int i = blockIdx.x * blockDim.x + threadIdx.x;
  if (i >= n) return;
  float v = A[i] + Bv[i];
  if (doRelu) v = fmaxf(v, 0.f);
  Y[i] = v;
}

__global__ void k_add_bias10(const float* A, const float* b, float* Y, int n10, int doRelu) {
  int i = blockIdx.x * blockDim.x + threadIdx.x;
  if (i >= n10) return;
  float v = A[i] + b[i % 10];
  if (doRelu) v = fmaxf(v, 0.f);
  Y[i] = v;
}

__global__ void k_div_cnt(float* agg, const float* cnt, int n10) {
  int i = blockIdx.x * blockDim.x + threadIdx.x;
  if (i >= n10) return;
  agg[i] /= fmaxf(cnt[i / 10], 1.f);
}

__global__ void k_concat26(const float* x, const float* x1, float* xc, int n26) {
  int i = blockIdx.x * blockDim.x + threadIdx.x;
  if (i >= n26) return;
  int r = i / 26, cv = i % 26;
  xc[i] = (cv < 16) ? x[(size_t)r * 16 + cv] : x1[(size_t)r * 10 + (cv - 16)];
}

// out[i] = softmax( [x3[p] || z[p]] @ W[20,4] + b )
__global__ void k_final(const float* __restrict__ x3, const float* __restrict__ zf,
                        const int* __restrict__ pick, const float* __restrict__ W,
                        const float* __restrict__ b, float* out, int npk) {
  int i = blockIdx.x * blockDim.x + threadIdx.x;
  if (i >= npk) return;
  int p = pick[i];
  float lg[4] = {b[0], b[1], b[2], b[3]};
#pragma unroll
  for (int d = 0; d < 10; ++d) {
    float xv = x3[(size_t)p * 10 + d];
    float zv = zf[(size_t)p * 10 + d];
#pragma unroll
    for (int o = 0; o < 4; ++o)
      lg[o] += xv * W[d * 4 + o] + zv * W[(10 + d) * 4 + o];
  }
  float m = fmaxf(fmaxf(lg[0], lg[1]), fmaxf(lg[2], lg[3]));
  float e0 = __expf(lg[0] - m), e1 = __expf(lg[1] - m),
        e2 = __expf(lg[2] - m), e3 = __expf(lg[3] - m);
  float s = e0 + e1 + e2 + e3;
  out[(size_t)i * 4 + 0] = e0 / s;
  out[(size_t)i * 4 + 1] = e1 / s;
  out[(size_t)i * 4 + 2] = e2 / s;
  out[(size_t)i * 4 + 3] = e3 / s;
}

// ======================================================================
// Host driver
// ======================================================================

// param leaf indices (pytree sorted-key order, after the 8 top-level arrays)
enum {
  IN_X = 0, IN_Z, IN_EI, IN_Z1, IN_Z2, IN_Z3, IN_EA, IN_PICK,
  P_ENC_W = 8, P_ENC_B,
  P_G2_WL, P_G2_WR, P_G2_ATT, P_G2_B, P_G2_BL, P_G2_BR,
  P_G3_WL, P_G3_WR, P_G3_ATT, P_G3_B, P_G3_BL, P_G3_BR,
  P_G5_WL, P_G5_WR, P_G5_ATT, P_G5_B, P_G5_BL, P_G5_BR,
  P_GC_WM, P_GC_BM,
  P_L1_WM, P_L1_WS, P_L1_BM,
  P_L2_WM, P_L2_WS, P_L2_BM,
  P_L3_WM, P_L3_WS, P_L3_BM,
  P_LIN_W, P_LIN_B,
  P_LXZ_W, P_LXZ_B,
  P_SG_WL, P_SG_WR, P_SG_BL,
  P_T1_WK, P_T1_WQ, P_T1_WS, P_T1_WV, P_T1_BK, P_T1_BQ, P_T1_BS, P_T1_BV,
  P_T2_WK, P_T2_WQ, P_T2_WS, P_T2_WV, P_T2_BK, P_T2_BQ, P_T2_BS, P_T2_BV,
};

extern "C" void kernel_launch(void* const* d_in, const int* in_sizes, int n_in,
                              void* d_out, int out_size, void* d_ws, size_t ws_size,
                              hipStream_t stream) {
  const int N   = in_sizes[IN_X] / 16;
  const int E   = in_sizes[IN_EI] / 2;
  const int NPK = in_sizes[IN_PICK];

  auto F = [&](int i) { return (const float*)d_in[i]; };
  auto I = [&](int i) { return (const int*)d_in[i]; };

  // workspace partition (floats)
  float* ws = (float*)d_ws;
  size_t off = 0;
  auto alloc = [&](size_t cnt) { float* p = ws + off; off += cnt; return p; };
  float* nodeA = alloc((size_t)N * 10);
  float* nodeB = alloc((size_t)N * 10);
  float* nodeC = alloc((size_t)N * 10);
  float* nodeD = alloc((size_t)N * 10);
  float* nodeE = alloc((size_t)N * 10);
  float* segM  = alloc((size_t)N);
  float* segS  = alloc((size_t)N);
  float* edgeW = alloc((size_t)E);
  float* xcat  = alloc((size_t)N * 26);
  (void)ws_size;

  const int N10 = N * 10;
  float encNegInfF;
  { unsigned u = ENC_NEG_INF_U; memcpy(&encNegInfF, &u, 4); }

  auto g1 = [](int n) { return dim3((unsigned)((n + 255) / 256)); };
  auto fill = [&](float* p, float v, int n) {
    k_fill<<<g1(n), 256, 0, stream>>>(p, v, n);
  };
  auto linear = [&](const float* X, const float* W, const float* B, const float* R,
                    float* Y, int inF, int outF, int relu) {
    int waves = (N + 15) / 16;
    k_linear_wmma<<<g1(waves * 32), 256, 0, stream>>>(X, W, B, R, Y, N, inF, outF, relu);
  };
  auto ownEdge = [&](const float* X, int inF, int wm, int bm, float* agg) {
    int waves = (E + 15) / 16;
    k_own_edge_wmma<<<g1(waves * 32), 256, 0, stream>>>(
        X, F(IN_EA), I(IN_EI), I(IN_EI) + E, F(wm), F(bm), agg, E, inF);
  };
  // full softmax aggregation given precomputed edge scores in edgeW
  auto softAgg = [&](const int* src, const int* tgt, const float* V, float* agg) {
    k_exp_sum<<<g1(E), 256, 0, stream>>>(segM, edgeW, segS, tgt, E);
    k_soft_scatter<<<g1(E), 256, 0, stream>>>(edgeW, segS, V, src, tgt, agg, E);
  };

  const int* s1 = I(IN_Z1);       const int* t1 = I(IN_Z1) + E;
  const int* s2 = I(IN_Z2);       const int* t2 = I(IN_Z2) + E;
  const int* s3 = I(IN_Z3);       const int* t3 = I(IN_Z3) + E;

  // ---------------- z stream ----------------
  // enc: z16 -> 10
  linear(F(IN_Z), F(P_ENC_W), F(P_ENC_B), nullptr, nodeA, 16, 10, 0);

  // tc1 (z1 edges), relu
  linear(nodeA, F(P_T1_WQ), F(P_T1_BQ), nullptr, nodeC, 10, 10, 0);  // q
  linear(nodeA, F(P_T1_WK), F(P_T1_BK), nullptr, nodeD, 10, 10, 0);  // k
  linear(nodeA, F(P_T1_WV), F(P_T1_BV), nullptr, nodeE, 10, 10, 0);  // v
  fill(segM, encNegInfF, N); fill(segS, 0.f, N); fill(nodeB, 0.f, N10);
  k_tc_edge<<<g1(E), 256, 0, stream>>>(nodeC, nodeD, s1, t1, edgeW, segM, E);
  softAgg(s1, t1, nodeE, nodeB);
  linear(nodeA, F(P_T1_WS), F(P_T1_BS), nodeB, nodeB, 10, 10, 1);    // agg + x@Ws + bs, relu

  // tc2 (z1 edges), no relu
  linear(nodeB, F(P_T2_WQ), F(P_T2_BQ), nullptr, nodeC, 10, 10, 0);
  linear(nodeB, F(P_T2_WK), F(P_T2_BK), nullptr, nodeD, 10, 10, 0);
  linear(nodeB, F(P_T2_WV), F(P_T2_BV), nullptr, nodeE, 10, 10, 0);
  fill(segM, encNegInfF, N); fill(segS, 0.f, N); fill(nodeA, 0.f, N10);
  k_tc_edge<<<g1(E), 256, 0, stream>>>(nodeC, nodeD, s1, t1, edgeW, segM, E);
  softAgg(s1, t1, nodeE, nodeA);
  linear(nodeB, F(P_T2_WS), F(P_T2_BS), nodeA, nodeA, 10, 10, 0);

  // linxz
  linear(nodeA, F(P_LXZ_W), F(P_LXZ_B), nullptr, nodeB, 10, 10, 0);

  // gc1 (z1 edges): relu(segsum(msg[src]) + z)
  linear(nodeB, F(P_GC_WM), F(P_GC_BM), nullptr, nodeC, 10, 10, 0);
  fill(nodeD, 0.f, N10);
  k_sum_scatter<<<g1(E), 256, 0, stream>>>(nodeC, s1, t1, nodeD, E);
  k_add_relu<<<g1(N10), 256, 0, stream>>>(nodeD, nodeB, nodeA, N10, 1);

  // gat2 (z2 edges), relu
  linear(nodeA, F(P_G2_WL), F(P_G2_BL), nullptr, nodeC, 10, 10, 0);  // xl
  linear(nodeA, F(P_G2_WR), F(P_G2_BR), nullptr, nodeD, 10, 10, 0);  // xr
  fill(segM, encNegInfF, N); fill(segS, 0.f, N); fill(nodeE, 0.f, N10);
  k_gat_edge<<<g1(E), 256, 0, stream>>>(nodeC, nodeD, F(P_G2_ATT), s2, t2, edgeW, segM, E);
  softAgg(s2, t2, nodeC, nodeE);
  k_add_bias10<<<g1(N10), 256, 0, stream>>>(nodeE, F(P_G2_B), nodeB, N10, 1);

  // gat3 (z1 edges), relu
  linear(nodeB, F(P_G3_WL), F(P_G3_BL), nullptr, nodeC, 10, 10, 0);
  linear(nodeB, F(P_G3_WR), F(P_G3_BR), nullptr, nodeD, 10, 10, 0);
  fill(segM, encNegInfF, N); fill(segS, 0.f, N); fill(nodeE, 0.f, N10);
  k_gat_edge<<<g1(E), 256, 0, stream>>>(nodeC, nodeD, F(P_G3_ATT), s1, t1, edgeW, segM, E);
  softAgg(s1, t1, nodeC, nodeE);
  k_add_bias10<<<g1(N10), 256, 0, stream>>>(nodeE, F(P_G3_B), nodeA, N10, 1);

  // sage4 (z3 edges), relu: relu(segmax(x)@Wl + bl + x@Wr)
  fill(nodeE, encNegInfF, N10);
  k_max_scatter<<<g1(E), 256, 0, stream>>>(nodeA, s3, t3, nodeE, E);
  k_dec_sanitize<<<g1(N10), 256, 0, stream>>>(nodeE, N10);
  linear(nodeE, F(P_SG_WL), F(P_SG_BL), nullptr, nodeB, 10, 10, 0);
  linear(nodeA, F(P_SG_WR), nullptr, nodeB, nodeB, 10, 10, 1);

  // gat5 (z1 edges), no relu -> z_final in nodeA
  linear(nodeB, F(P_G5_WL), F(P_G5_BL), nullptr, nodeC, 10, 10, 0);
  linear(nodeB, F(P_G5_WR), F(P_G5_BR), nullptr, nodeD, 10, 10, 0);
  fill(segM, encNegInfF, N); fill(segS, 0.f, N); fill(nodeE, 0.f, N10);
  k_gat_edge<<<g1(E), 256, 0, stream>>>(nodeC, nodeD, F(P_G5_ATT), s1, t1, edgeW, segM, E);
  softAgg(s1, t1, nodeC, nodeE);
  k_add_bias10<<<g1(N10), 256, 0, stream>>>(nodeE, F(P_G5_B), nodeA, N10, 0);

  // ---------------- x stream (edge_index, edge_attr) ----------------
  fill(segS, 0.f, N);
  k_deg<<<g1(E), 256, 0, stream>>>(I(IN_EI) + E, segS, E);  // in-degree counts

  // l1: x[N,16] -> x1[N,10] in nodeD
  fill(nodeC, 0.f, N10);
  ownEdge(F(IN_X), 16, P_L1_WM, P_L1_BM, nodeC);
  k_div_cnt<<<g1(N10), 256, 0, stream>>>(nodeC, segS, N10);
  linear(F(IN_X), F(P_L1_WS), nullptr, nodeC, nodeD, 16, 10, 0);

  // xcat = [x || x1] [N,26]
  k_concat26<<<g1(N * 26), 256, 0, stream>>>(F(IN_X), nodeD, xcat, N * 26);

  // l2: xcat[N,26] -> nodeD
  fill(nodeC, 0.f, N10);
  ownEdge(xcat, 26, P_L2_WM, P_L2_BM, nodeC);
  k_div_cnt<<<g1(N10), 256, 0, stream>>>(nodeC, segS, N10);
  linear(xcat, F(P_L2_WS), nullptr, nodeC, nodeD, 26, 10, 0);

  // l3: nodeD[N,10] -> nodeE (x3)
  fill(nodeC, 0.f, N10);
  ownEdge(nodeD, 10, P_L3_WM, P_L3_BM, nodeC);
  k_div_cnt<<<g1(N10), 256, 0, stream>>>(nodeC, segS, N10);
  linear(nodeD, F(P_L3_WS), nullptr, nodeC, nodeE, 10, 10, 0);

  // final: softmax(([x3 || z])[pickable] @ W20x4 + b)
  k_final<<<g1(NPK), 256, 0, stream>>>(nodeE, nodeA, I(IN_PICK),
                                       F(P_LIN_W), F(P_LIN_B), (float*)d_out, NPK);
  (void)n_in; (void)out_size;
}